// BigGNN_87823491268665
// MI455X (gfx1250) — compile-verified
//
#include <hip/hip_runtime.h>
#include <math.h>

typedef float v2f __attribute__((ext_vector_type(2)));
typedef float v8f __attribute__((ext_vector_type(8)));

constexpr int   kD  = 300;
constexpr int   kH  = 2;
constexpr int   kHD = 600;   // H * D
constexpr int   kN1 = 384;
constexpr int   kN2 = 384;
constexpr int   kNC = 768;   // N1 + N2
constexpr int   kE  = 6144;
constexpr int   kL  = 3;
constexpr float kScale = 0.05773502691896258f;  // 1/sqrt(300)

__device__ __forceinline__ float leakyf(float x) { return x > 0.f ? x : 0.01f * x; }

__device__ __forceinline__ float wred_sum(float v) {
  for (int off = 16; off; off >>= 1) v += __shfl_xor(v, off, 32);
  return v;
}
__device__ __forceinline__ float wred_max(float v) {
  for (int off = 16; off; off >>= 1) v = fmaxf(v, __shfl_xor(v, off, 32));
  return v;
}

__device__ void atomicMaxF(float* addr, float val) {
  unsigned int* ua = (unsigned int*)addr;
  unsigned int old = *ua;
  while (__uint_as_float(old) < val) {
    unsigned int assumed = old;
    old = atomicCAS(ua, assumed, __float_as_uint(val));
    if (old == assumed) break;
  }
}

// ---------------------------------------------------------------------------
// WMMA fp32 GEMM: C[M,N] = scale * (A[M,K] @ op(B)) + bias
//   TRANSB == 0 : B is K x N row-major
//   TRANSB == 1 : B is N x K row-major (C = A @ B^T), for Q @ K^T
// One wave computes a 16(M) x 64(N) strip: 4 accumulators, each K-step does
// one A-fragment load reused by 4 B-fragment loads -> 4 v_wmma_f32_16x16x4.
// Branchless N-edge handling via clamped index + multiplicative mask.
// M must be a multiple of 16; K a multiple of 4.
// ---------------------------------------------------------------------------
template <int TRANSB>
__global__ void gemm_wmma(const float* __restrict__ A, int lda,
                          const float* __restrict__ B, int ldb,
                          float* __restrict__ C, int ldc,
                          int N, int K,
                          const float* __restrict__ bias,
                          float scale) {
  const int tm   = blockIdx.x;
  const int tn   = blockIdx.y;
  const int lane = threadIdx.x & 31;
  const int half = lane >> 4;    // 0: K pair {0,1}, 1: K pair {2,3}
  const int r16  = lane & 15;
  const int arow = tm * 16 + r16;

  // Per-sub-tile column, clamped to stay in-bounds, with 0/1 mask.
  int   col[4];
  float msk[4];
#pragma unroll
  for (int t = 0; t < 4; ++t) {
    const int bc = tn * 64 + t * 16 + r16;
    msk[t] = (bc < N) ? 1.f : 0.f;
    col[t] = (bc < N) ? bc : (N - 1);
  }

  const float* Ap = A + (size_t)arow * lda + 2 * half;

  v8f c0 = {}, c1 = {}, c2 = {}, c3 = {};
  for (int k0 = 0; k0 < K; k0 += 4) {
    const int ka = k0 + 2 * half;
    v2f a;
    a.x = Ap[k0];
    a.y = Ap[k0 + 1];
    v2f b0, b1, b2, b3;
    if (TRANSB) {
      b0.x = B[(size_t)col[0] * ldb + ka] * msk[0];
      b0.y = B[(size_t)col[0] * ldb + ka + 1] * msk[0];
      b1.x = B[(size_t)col[1] * ldb + ka] * msk[1];
      b1.y = B[(size_t)col[1] * ldb + ka + 1] * msk[1];
      b2.x = B[(size_t)col[2] * ldb + ka] * msk[2];
      b2.y = B[(size_t)col[2] * ldb + ka + 1] * msk[2];
      b3.x = B[(size_t)col[3] * ldb + ka] * msk[3];
      b3.y = B[(size_t)col[3] * ldb + ka + 1] * msk[3];
    } else {
      const float* Br0 = B + (size_t)ka * ldb;
      const float* Br1 = B + (size_t)(ka + 1) * ldb;
      b0.x = Br0[col[0]] * msk[0];  b0.y = Br1[col[0]] * msk[0];
      b1.x = Br0[col[1]] * msk[1];  b1.y = Br1[col[1]] * msk[1];
      b2.x = Br0[col[2]] * msk[2];  b2.y = Br1[col[2]] * msk[2];
      b3.x = Br0[col[3]] * msk[3];  b3.y = Br1[col[3]] * msk[3];
    }
    c0 = __builtin_amdgcn_wmma_f32_16x16x4_f32(false, a, false, b0, (short)0, c0, false, false);
    c1 = __builtin_amdgcn_wmma_f32_16x16x4_f32(false, a, false, b1, (short)0, c1, false, false);
    c2 = __builtin_amdgcn_wmma_f32_16x16x4_f32(false, a, false, b2, (short)0, c2, false, false);
    c3 = __builtin_amdgcn_wmma_f32_16x16x4_f32(false, a, false, b3, (short)0, c3, false, false);
  }

  v8f acc[4] = {c0, c1, c2, c3};
#pragma unroll
  for (int t = 0; t < 4; ++t) {
    const int bc = tn * 64 + t * 16 + r16;
    if (bc < N) {
      const float bb = bias ? bias[bc] : 0.f;
#pragma unroll
      for (int r = 0; r < 8; ++r) {
        const int row = tm * 16 + r + 8 * half;
        C[(size_t)row * ldc + bc] = acc[t][r] * scale + bb;
      }
    }
  }
}

// ---------------------------------------------------------------------------
// Utility kernels
// ---------------------------------------------------------------------------
__global__ void fill_kernel(float* __restrict__ p, float val, int n) {
  int i = blockIdx.x * blockDim.x + threadIdx.x;
  if (i < n) p[i] = val;
}
__global__ void copy_kernel(float* __restrict__ dst, const float* __restrict__ src, int n) {
  int i = blockIdx.x * blockDim.x + threadIdx.x;
  if (i < n) dst[i] = src[i];
}

// ---------------------------------------------------------------------------
// Sparse (edge-list) attention pieces
// ---------------------------------------------------------------------------
__global__ void edge_logits_kernel(const float* __restrict__ q, const float* __restrict__ k,
                                   const int* __restrict__ src, const int* __restrict__ dst,
                                   float* __restrict__ lg) {
  const int wid  = (blockIdx.x * blockDim.x + threadIdx.x) >> 5;
  const int lane = threadIdx.x & 31;
  if (wid >= kE * kH) return;
  const int e = wid >> 1, h = wid & 1;
  const float* qp = q + (size_t)dst[e] * kHD + h * kD;
  const float* kp = k + (size_t)src[e] * kHD + h * kD;
  float acc = 0.f;
  for (int i = lane; i < kD; i += 32) acc += qp[i] * kp[i];
  acc = wred_sum(acc);
  if (lane == 0) lg[wid] = acc * kScale;
}

__global__ void edge_max_kernel(const float* __restrict__ lg, const int* __restrict__ dst,
                                float* __restrict__ m) {
  int t = blockIdx.x * blockDim.x + threadIdx.x;
  if (t >= kE * kH) return;
  const int e = t >> 1, h = t & 1;
  atomicMaxF(&m[dst[e] * kH + h], lg[t]);
}

__global__ void m_fix_kernel(float* __restrict__ m, int n) {
  int t = blockIdx.x * blockDim.x + threadIdx.x;
  if (t >= n) return;
  float v = m[t];
  if (!(v > -INFINITY) || !(v < INFINITY)) m[t] = 0.f;
}

__global__ void edge_expsum_kernel(const float* __restrict__ lg, const int* __restrict__ dst,
                                   const float* __restrict__ m, float* __restrict__ eb,
                                   float* __restrict__ s) {
  int t = blockIdx.x * blockDim.x + threadIdx.x;
  if (t >= kE * kH) return;
  const int e = t >> 1, h = t & 1;
  float v = expf(lg[t] - m[dst[e] * kH + h]);
  eb[t] = v;
  atomicAdd(&s[dst[e] * kH + h], v);
}

__global__ void edge_scatter_kernel(const float* __restrict__ eb, const float* __restrict__ s,
                                    const float* __restrict__ v,
                                    const int* __restrict__ src, const int* __restrict__ dst,
                                    float* __restrict__ attn) {
  int t = blockIdx.x * blockDim.x + threadIdx.x;
  if (t >= kE * kH * kD) return;
  const int d  = t % kD;
  const int eh = t / kD;
  const int e  = eh >> 1, h = eh & 1;
  const int dn = dst[e], sn = src[e];
  const float alpha = eb[eh] / fmaxf(s[dn * kH + h], 1e-16f);
  atomicAdd(&attn[(size_t)dn * kHD + h * kD + d],
            v[(size_t)sn * kHD + h * kD + d] * alpha);
}

__global__ void finalize_kernel(const float* __restrict__ attn, const float* __restrict__ skip,
                                float* __restrict__ xout, int n) {
  int t = blockIdx.x * blockDim.x + threadIdx.x;
  if (t >= n * kD) return;
  const int i = t / kD, d = t % kD;
  float o = 0.5f * (attn[(size_t)i * kHD + d] + attn[(size_t)i * kHD + kD + d]) + skip[t];
  xout[t] = leakyf(o);
}

// ---------------------------------------------------------------------------
// Dense row softmax (one wave per row), in-place
// ---------------------------------------------------------------------------
__global__ void softmax_rows(float* __restrict__ P, int rows, int cols) {
  const int wid  = (blockIdx.x * blockDim.x + threadIdx.x) >> 5;
  const int lane = threadIdx.x & 31;
  if (wid >= rows) return;
  float* p = P + (size_t)wid * cols;
  float mx = -INFINITY;
  for (int i = lane; i < cols; i += 32) mx = fmaxf(mx, p[i]);
  mx = wred_max(mx);
  float sm = 0.f;
  for (int i = lane; i < cols; i += 32) { float e = expf(p[i] - mx); p[i] = e; sm += e; }
  sm = wred_sum(sm);
  const float inv = 1.f / fmaxf(sm, 1e-16f);
  for (int i = lane; i < cols; i += 32) p[i] *= inv;
}

// ---------------------------------------------------------------------------
// Readout head
// ---------------------------------------------------------------------------
__global__ void mean_kernel(const float* __restrict__ x, int n, float* __restrict__ out) {
  int j = blockIdx.x * blockDim.x + threadIdx.x;
  if (j >= kD) return;
  float acc = 0.f;
  for (int i = 0; i < n; ++i) acc += x[(size_t)i * kD + j];
  out[j] = acc / (float)n;
}

__global__ void mlp1_kernel(const float* __restrict__ p12, const float* __restrict__ W1,
                            const float* __restrict__ b1, float* __restrict__ h1) {
  int j = blockIdx.x * blockDim.x + threadIdx.x;
  if (j >= kD) return;
  float acc = b1[j];
  for (int t = 0; t < 2 * kD; ++t) acc += p12[t] * W1[(size_t)t * kD + j];
  h1[j] = leakyf(acc);
}

__global__ void mlp2_kernel(const float* __restrict__ h1, const float* __restrict__ W2,
                            const float* __restrict__ b2, float* __restrict__ h2) {
  int j = blockIdx.x * blockDim.x + threadIdx.x;
  if (j >= kD) return;
  float acc = b2[j];
  for (int t = 0; t < kD; ++t) acc += h1[t] * W2[(size_t)t * kD + j];
  h2[j] = leakyf(acc);
}

__global__ void mlp3_kernel(const float* __restrict__ h2, const float* __restrict__ W3,
                            const float* __restrict__ b3, float* __restrict__ out) {
  const int lane = threadIdx.x & 31;
  float acc = 0.f;
  for (int i = lane; i < kD; i += 32) acc += h2[i] * W3[i];
  acc = wred_sum(acc);
  if (threadIdx.x == 0) out[0] = 1.f / (1.f + expf(-(acc + b3[0])));
}

// ---------------------------------------------------------------------------
extern "C" void kernel_launch(void* const* d_in, const int* in_sizes, int n_in,
                              void* d_out, int out_size, void* d_ws, size_t ws_size,
                              hipStream_t stream) {
  (void)in_sizes; (void)n_in; (void)out_size; (void)ws_size;

  const float* in_x1 = (const float*)d_in[0];
  const float* in_x2 = (const float*)d_in[1];
  const int*   ei1   = (const int*)d_in[2];
  const int*   ei2   = (const int*)d_in[3];
  // d_in[4], d_in[5] = edge_attr (unused by the reference)
  float* out = (float*)d_out;

  // ---- workspace carve-up (fp32) ----
  float* ws = (float*)d_ws;
  size_t off = 0;
  auto WA = [&](size_t n) { float* p = ws + off; off += n; return p; };
  float* x1   = WA((size_t)kN1 * kD);
  float* x2   = WA((size_t)kN2 * kD);
  float* x1n  = WA((size_t)kN1 * kD);
  float* x2n  = WA((size_t)kN2 * kD);
  float* xc1  = WA((size_t)kNC * kD);
  float* xc2  = WA((size_t)kNC * kD);
  float* xc1n = WA((size_t)kN1 * kD);
  float* xc2n = WA((size_t)kN2 * kD);
  float* qb   = WA((size_t)kNC * kHD);
  float* kb   = WA((size_t)kNC * kHD);
  float* vb   = WA((size_t)kNC * kHD);
  float* skip = WA((size_t)kNC * kD);
  float* lg   = WA((size_t)kE * kH);
  float* eb   = WA((size_t)kE * kH);
  float* mb   = WA((size_t)kNC * kH);
  float* sb   = WA((size_t)kNC * kH);
  float* attn = WA((size_t)kNC * kHD);
  float* plg  = WA((size_t)kH * kN1 * kN1);
  float* h1b  = WA(kD);
  float* h2b  = WA(kD);

  auto cdiv = [](int a, int b) { return (a + b - 1) / b; };

  auto gemm_nn = [&](const float* A, int lda, const float* B, int ldb, float* C, int ldc,
                     int M, int N, int K, const float* bias, float scale) {
    dim3 g(M / 16, cdiv(N, 64));
    gemm_wmma<0><<<g, dim3(32), 0, stream>>>(A, lda, B, ldb, C, ldc, N, K, bias, scale);
  };
  auto gemm_nt = [&](const float* A, int lda, const float* B, int ldb, float* C, int ldc,
                     int M, int N, int K, const float* bias, float scale) {
    dim3 g(M / 16, cdiv(N, 64));
    gemm_wmma<1><<<g, dim3(32), 0, stream>>>(A, lda, B, ldb, C, ldc, N, K, bias, scale);
  };
  auto fillb = [&](float* p, float v, int n) {
    fill_kernel<<<cdiv(n, 256), 256, 0, stream>>>(p, v, n);
  };
  auto copyb = [&](float* d, const float* s, int n) {
    copy_kernel<<<cdiv(n, 256), 256, 0, stream>>>(d, s, n);
  };

  struct WG { const float *Wq, *bq, *Wk, *bk, *Wv, *bv, *Ws, *bs; };
  auto group = [&](int base, int l) {
    WG g;
    g.Wq = (const float*)d_in[base + 0] + (size_t)l * kD * kHD;
    g.bq = (const float*)d_in[base + 1] + (size_t)l * kHD;
    g.Wk = (const float*)d_in[base + 2] + (size_t)l * kD * kHD;
    g.bk = (const float*)d_in[base + 3] + (size_t)l * kHD;
    g.Wv = (const float*)d_in[base + 4] + (size_t)l * kD * kHD;
    g.bv = (const float*)d_in[base + 5] + (size_t)l * kHD;
    g.Ws = (const float*)d_in[base + 6] + (size_t)l * kD * kD;
    g.bs = (const float*)d_in[base + 7] + (size_t)l * kD;
    return g;
  };

  // ---- sparse (edge-list) transformer conv: n==384 nodes ----
  auto self_tconv = [&](const float* x, int n, const int* ei, const WG& g, float* xout) {
    gemm_nn(x, kD, g.Wq, kHD, qb, kHD, n, kHD, kD, g.bq, 1.f);
    gemm_nn(x, kD, g.Wk, kHD, kb, kHD, n, kHD, kD, g.bk, 1.f);
    gemm_nn(x, kD, g.Wv, kHD, vb, kHD, n, kHD, kD, g.bv, 1.f);
    gemm_nn(x, kD, g.Ws, kD, skip, kD, n, kD, kD, g.bs, 1.f);
    const int* src = ei;
    const int* dst = ei + kE;
    fillb(mb, -INFINITY, n * kH);
    fillb(sb, 0.f, n * kH);
    fillb(attn, 0.f, n * kHD);
    edge_logits_kernel<<<cdiv(kE * kH * 32, 256), 256, 0, stream>>>(qb, kb, src, dst, lg);
    edge_max_kernel<<<cdiv(kE * kH, 256), 256, 0, stream>>>(lg, dst, mb);
    m_fix_kernel<<<cdiv(n * kH, 256), 256, 0, stream>>>(mb, n * kH);
    edge_expsum_kernel<<<cdiv(kE * kH, 256), 256, 0, stream>>>(lg, dst, mb, eb, sb);
    edge_scatter_kernel<<<cdiv(kE * kH * kD, 256), 256, 0, stream>>>(eb, sb, vb, src, dst, attn);
    finalize_kernel<<<cdiv(n * kD, 256), 256, 0, stream>>>(attn, skip, xout, n);
  };

  // ---- dense bipartite cross attention: dst rows [0,384), src rows [384,768) ----
  auto cross_tconv = [&](const float* xc, const WG& g, float* xout) {
    gemm_nn(xc, kD, g.Wq, kHD, qb, kHD, kN1, kHD, kD, g.bq, 1.f);                    // Q of dst
    gemm_nn(xc + (size_t)kN1 * kD, kD, g.Wk, kHD, kb, kHD, kN2, kHD, kD, g.bk, 1.f); // K of src
    gemm_nn(xc + (size_t)kN1 * kD, kD, g.Wv, kHD, vb, kHD, kN2, kHD, kD, g.bv, 1.f); // V of src
    gemm_nn(xc, kD, g.Ws, kD, skip, kD, kN1, kD, kD, g.bs, 1.f);
    for (int h = 0; h < kH; ++h) {
      float* P = plg + (size_t)h * kN1 * kN1;
      gemm_nt(qb + h * kD, kHD, kb + h * kD, kHD, P, kN1, kN1, kN1, kD, nullptr, kScale);
      softmax_rows<<<cdiv(kN1 * 32, 256), 256, 0, stream>>>(P, kN1, kN1);
      gemm_nn(P, kN1, vb + h * kD, kHD, attn + h * kD, kHD, kN1, kD, kN1, nullptr, 1.f);
    }
    finalize_kernel<<<cdiv(kN1 * kD, 256), 256, 0, stream>>>(attn, skip, xout, kN1);
  };

  // ---- forward pass ----
  copyb(x1, in_x1, kN1 * kD);
  copyb(x2, in_x2, kN2 * kD);

  for (int l = 0; l < kL; ++l) {
    const WG tsa = group(6, l), gsa = group(14, l), tca = group(22, l), gca = group(30, l);
    self_tconv(x1, kN1, ei1, tsa, x1n);
    self_tconv(x2, kN2, ei2, gsa, x2n);
    copyb(xc1, x1n, kN1 * kD);
    copyb(xc1 + (size_t)kN1 * kD, x2n, kN2 * kD);
    copyb(xc2, x2n, kN2 * kD);
    copyb(xc2 + (size_t)kN2 * kD, x1n, kN1 * kD);
    cross_tconv(xc1, tca, xc1n);
    cross_tconv(xc2, gca, xc2n);
    copyb(x1, xc1n, kN1 * kD);
    copyb(x2, xc2n, kN2 * kD);
  }

  // ---- readout: p1, p2, sigmoid(mlp(concat(p1,p2))) ----
  mean_kernel<<<cdiv(kD, 256), 256, 0, stream>>>(x1, kN1, out);        // p1
  mean_kernel<<<cdiv(kD, 256), 256, 0, stream>>>(x2, kN2, out + kD);   // p2
  mlp1_kernel<<<cdiv(kD, 256), 256, 0, stream>>>(out, (const float*)d_in[38],
                                                 (const float*)d_in[39], h1b);
  mlp2_kernel<<<cdiv(kD, 256), 256, 0, stream>>>(h1b, (const float*)d_in[40],
                                                 (const float*)d_in[41], h2b);
  mlp3_kernel<<<1, 32, 0, stream>>>(h2b, (const float*)d_in[42],
                                    (const float*)d_in[43], out + 2 * kD);
}